// GNN_11879879541058
// MI455X (gfx1250) — compile-verified
//
#include <hip/hip_runtime.h>
#include <hip/hip_bf16.h>

// ---------------------------------------------------------------------------
// GCN forward on gfx1250: fp32 WMMA (V_WMMA_F32_16X16X4_F32) for the dense
// feature transforms, float atomics for the edge scatter (the real bottleneck).
// ---------------------------------------------------------------------------

typedef __attribute__((ext_vector_type(2))) float v2f;
typedef __attribute__((ext_vector_type(8))) float v8f;

#define GNN_N 100000
#define GNN_E 1000000
#define GNN_H 64
#define GNN_G 128

// ---------------- generic fill ----------------
__global__ void fill_kernel(float* __restrict__ p, float v, int n) {
    int i = blockIdx.x * blockDim.x + threadIdx.x;
    if (i < n) p[i] = v;
}

// ---------------- degree accumulation (dst side, self-loop pre-seeded) ------
__global__ void deg_accum_kernel(const int* __restrict__ dst,
                                 float* __restrict__ deg, int e) {
    int i = blockIdx.x * blockDim.x + threadIdx.x;
    if (i < e) atomicAdd(&deg[dst[i]], 1.0f);
}

__global__ void dinv_kernel(const float* __restrict__ deg,
                            float* __restrict__ dinv, int n) {
    int i = blockIdx.x * blockDim.x + threadIdx.x;
    if (i < n) dinv[i] = rsqrtf(deg[i]);   // deg >= 1 always (self-loop)
}

// ---------------- fp32 WMMA GEMM:  Hout[N,64] = X[N,64] @ W[64,64] ----------
// One wave computes one 16x16 output tile via 16 chained 16x16x4 f32 WMMAs.
// Block = 256 threads = 8 waves = 32 rows x 64 cols. N % 32 == 0 -> EXEC all 1s.
__global__ __launch_bounds__(256) void gemm64_wmma_kernel(
        const float* __restrict__ X, const float* __restrict__ W,
        float* __restrict__ Hout, int n) {
    const int wave = threadIdx.x >> 5;          // 0..7
    const int lane = threadIdx.x & 31;
    const int rowBase = (blockIdx.x * 2 + (wave >> 2)) * 16;  // 2 row tiles/block
    const int colBase = (wave & 3) * 16;                      // 4 col tiles
    if (rowBase >= n) return;                   // wave-uniform

    const int m    = lane & 15;                 // A row within tile / B,C column
    const int half = lane >> 4;                 // lane half selects K/M pairs
    const int koff = half * 2;                  // A: lanes16-31 hold K=2,3

    v8f c = {};
    #pragma unroll
    for (int k0 = 0; k0 < GNN_H; k0 += 4) {
        // A fragment: 16x4 tile of X, 2 consecutive K values per lane
        v2f a = *reinterpret_cast<const v2f*>(&X[(rowBase + m) * GNN_H + k0 + koff]);
        // B fragment: 4x16 tile of W; VGPR v holds row K = k0 + v + 2*half
        v2f b;
        b.x = W[(k0 + 0 + 2 * half) * GNN_H + colBase + m];
        b.y = W[(k0 + 1 + 2 * half) * GNN_H + colBase + m];
        c = __builtin_amdgcn_wmma_f32_16x16x4_f32(
                /*neg_a=*/false, a, /*neg_b=*/false, b,
                /*c_mod=*/(short)0, c, /*reuse_a=*/false, /*reuse_b=*/false);
    }

    const int col = colBase + m;
    #pragma unroll
    for (int v = 0; v < 8; ++v) {               // C/D: row = v + 8*half
        Hout[(rowBase + v + 8 * half) * GNN_H + col] = c[v];
    }
}

// ---------------- edge scatter: Out[dst] += H[src] * dinv[src]*dinv[dst] ----
// 16 lanes per edge, each lane handles 4 channels (float4 gather, 4 atomics).
__global__ void edge_scatter_kernel(const int* __restrict__ src,
                                    const int* __restrict__ dst,
                                    const float* __restrict__ dinv,
                                    const float* __restrict__ H,
                                    float* __restrict__ Out, int e) {
    int t = blockIdx.x * blockDim.x + threadIdx.x;
    int edge = t >> 4;
    if (edge >= e) return;
    int c4 = (t & 15) * 4;
    int s = src[edge];
    int d = dst[edge];
    float norm = dinv[s] * dinv[d];
    const float4 hv = *reinterpret_cast<const float4*>(H + s * GNN_H + c4);
    float* o = Out + d * GNN_H + c4;
    atomicAdd(o + 0, hv.x * norm);
    atomicAdd(o + 1, hv.y * norm);
    atomicAdd(o + 2, hv.z * norm);
    atomicAdd(o + 3, hv.w * norm);
}

// ---------------- self-loop + bias + ReLU ----------------
__global__ void finalize_kernel(const float* __restrict__ Hself,
                                const float* __restrict__ dinv,
                                const float* __restrict__ bias,
                                float* __restrict__ Out, int n) {
    int t = blockIdx.x * blockDim.x + threadIdx.x;
    if (t >= n * GNN_H) return;
    int i = t >> 6;
    int c = t & 63;
    float di = dinv[i];
    float v = Out[t] + Hself[t] * di * di + bias[c];
    Out[t] = v > 0.0f ? v : 0.0f;
}

// ---------------- mean pool accumulation ----------------
__global__ void pool_accum_kernel(const float* __restrict__ H,
                                  const int* __restrict__ batch,
                                  float* __restrict__ pool,
                                  float* __restrict__ cnt, int n) {
    int t = blockIdx.x * blockDim.x + threadIdx.x;
    if (t >= n * GNN_H) return;
    int i = t >> 6;
    int c = t & 63;
    int g = batch[i];
    atomicAdd(&pool[g * GNN_H + c], H[t]);
    if (c == 0) atomicAdd(&cnt[g], 1.0f);
}

// ---------------- readout head: out[g] = mean(pool[g]) @ W_out + b_out -----
__global__ void head_kernel(const float* __restrict__ pool,
                            const float* __restrict__ cnt,
                            const float* __restrict__ Wout,
                            const float* __restrict__ bout,
                            float* __restrict__ out) {
    int g = blockIdx.x * blockDim.x + threadIdx.x;
    if (g >= GNN_G) return;
    float inv = 1.0f / fmaxf(cnt[g], 1.0f);
    float acc = 0.0f;
    #pragma unroll
    for (int c = 0; c < GNN_H; ++c)
        acc += pool[g * GNN_H + c] * inv * Wout[c];
    out[g] = acc + bout[0];
}

// ---------------------------------------------------------------------------
extern "C" void kernel_launch(void* const* d_in, const int* in_sizes, int n_in,
                              void* d_out, int out_size, void* d_ws, size_t ws_size,
                              hipStream_t stream) {
    (void)in_sizes; (void)n_in; (void)out_size; (void)ws_size;

    const float* x     = (const float*)d_in[0];
    const int*   eidx  = (const int*)d_in[1];      // [2, E] flat: src row, dst row
    const int*   batch = (const int*)d_in[2];
    const float* W1    = (const float*)d_in[3];
    const float* b1    = (const float*)d_in[4];
    const float* W2    = (const float*)d_in[5];
    const float* b2    = (const float*)d_in[6];
    const float* Wout  = (const float*)d_in[7];
    const float* bout  = (const float*)d_in[8];
    float* out = (float*)d_out;

    const int* src = eidx;
    const int* dst = eidx + GNN_E;

    // workspace layout (floats)
    float* ws   = (float*)d_ws;
    float* deg  = ws;                                   // N
    float* dinv = deg  + GNN_N;                         // N
    float* bufA = dinv + GNN_N;                         // N*64 (GEMM output h)
    float* bufB = bufA + (size_t)GNN_N * GNN_H;         // N*64 (aggregation out)
    float* pool = bufB + (size_t)GNN_N * GNN_H;         // G*64
    float* cnt  = pool + (size_t)GNN_G * GNN_H;         // G

    const int TPB = 256;
    auto blocks = [](long long n, int tpb) { return (int)((n + tpb - 1) / tpb); };

    const long long NH = (long long)GNN_N * GNN_H;

    // --- degree / normalization ---
    fill_kernel<<<blocks(GNN_N, TPB), TPB, 0, stream>>>(deg, 1.0f, GNN_N); // self-loops
    deg_accum_kernel<<<blocks(GNN_E, TPB), TPB, 0, stream>>>(dst, deg, GNN_E);
    dinv_kernel<<<blocks(GNN_N, TPB), TPB, 0, stream>>>(deg, dinv, GNN_N);

    // --- layer 1: h1 = x @ W1 ; agg ; +self +b1 ; ReLU ---
    gemm64_wmma_kernel<<<GNN_N / 32, TPB, 0, stream>>>(x, W1, bufA, GNN_N);
    fill_kernel<<<blocks(NH, TPB), TPB, 0, stream>>>(bufB, 0.0f, (int)NH);
    edge_scatter_kernel<<<blocks((long long)GNN_E * 16, TPB), TPB, 0, stream>>>(
        src, dst, dinv, bufA, bufB, GNN_E);
    finalize_kernel<<<blocks(NH, TPB), TPB, 0, stream>>>(bufA, dinv, b1, bufB, GNN_N);

    // --- layer 2: h2 = out1 @ W2 ; agg ; +self +b2 ; ReLU ---
    gemm64_wmma_kernel<<<GNN_N / 32, TPB, 0, stream>>>(bufB, W2, bufA, GNN_N);
    fill_kernel<<<blocks(NH, TPB), TPB, 0, stream>>>(bufB, 0.0f, (int)NH);
    edge_scatter_kernel<<<blocks((long long)GNN_E * 16, TPB), TPB, 0, stream>>>(
        src, dst, dinv, bufA, bufB, GNN_E);
    finalize_kernel<<<blocks(NH, TPB), TPB, 0, stream>>>(bufA, dinv, b2, bufB, GNN_N);

    // --- global mean pool + linear head ---
    fill_kernel<<<blocks(GNN_G * GNN_H, TPB), TPB, 0, stream>>>(pool, 0.0f, GNN_G * GNN_H);
    fill_kernel<<<1, TPB, 0, stream>>>(cnt, 0.0f, GNN_G);
    pool_accum_kernel<<<blocks(NH, TPB), TPB, 0, stream>>>(bufB, batch, pool, cnt, GNN_N);
    head_kernel<<<1, GNN_G, 0, stream>>>(pool, cnt, Wout, bout, out);
}